// GATConv_54279796687119
// MI455X (gfx1250) — compile-verified
//
#include <hip/hip_runtime.h>
#include <hip/hip_bf16.h>
#include <math.h>

// ---------------------------------------------------------------------------
// GAT forward for MI455X (gfx1250), wave32.
//  K1: xp = x @ W           via v_wmma_f32_16x16x4_f32 (fp32 WMMA, exact)
//  K2: per-(node,head) attention dot products (C=32)
//  K3: init out=bias, seg_max=-inf, seg_sum=0
//  K4: edge logits + leakyReLU + float atomic max  (segment_max)
//  K5: exp(logit - max) + float atomic add         (segment_sum)
//  K6: alpha = e/sum, write alpha output, scatter-add alpha*xp[src] into out
// ---------------------------------------------------------------------------

typedef __attribute__((ext_vector_type(2))) float v2f;
typedef __attribute__((ext_vector_type(8))) float v8f;

#define HEADS 4
#define CH    32
#define HC    128   // HEADS*CH == F
#define LEAKY 0.2f

// ---------------- K1: WMMA GEMM  xp[n, hc] = sum_f x[n,f] * W[f,hc] ----------
// One wave -> one 16x16 tile of xp. K=128 consumed in 32 steps of 4.
__global__ void __launch_bounds__(128)
gat_gemm_wmma(const float* __restrict__ x, const float* __restrict__ W,
              float* __restrict__ xp, int n) {
    const int wave = threadIdx.x >> 5;            // 0..3
    const int lane = threadIdx.x & 31;
    const int tile = blockIdx.x * 4 + wave;
    const int tilesN = HC / 16;                   // 8
    const int tm = tile / tilesN;                 // row tile
    const int tn = tile % tilesN;                 // col tile
    if (tm * 16 >= n) return;

    const int r     = lane & 15;
    const int kbase = (lane >> 4) << 1;           // 0 or 2

    int arow = tm * 16 + r;
    if (arow >= n) arow = n - 1;                  // defensive clamp (n%16==0 normally)
    const float* __restrict__ xrow = x + (size_t)arow * HC;
    const float* __restrict__ wcol = W + tn * 16 + r;   // stride HC down K

    v8f acc = {};
#pragma unroll
    for (int kk = 0; kk < HC; kk += 4) {
        const int k = kk + kbase;
        v2f a, b;
        // A: 16x4 tile, lane holds A[M=r][k], A[M=r][k+1]
        const float2 av = *(const float2*)(xrow + k);
        a.x = av.x; a.y = av.y;
        // B: 4x16 tile, lane holds B[k][N=r], B[k+1][N=r]
        b.x = wcol[(size_t)k * HC];
        b.y = wcol[(size_t)(k + 1) * HC];
        acc = __builtin_amdgcn_wmma_f32_16x16x4_f32(
            /*neg_a=*/false, a, /*neg_b=*/false, b,
            /*c_mod=*/(short)0, acc, /*reuse_a=*/false, /*reuse_b=*/false);
    }

    // D layout: VGPR i -> lanes 0-15: D[M=i][N=lane]; lanes 16-31: D[M=8+i][N=lane-16]
    const int mbase = tm * 16 + ((lane >> 4) << 3);
    const int col   = tn * 16 + (lane & 15);
#pragma unroll
    for (int i = 0; i < 8; ++i) {
        const int row = mbase + i;
        if (row < n) xp[(size_t)row * HC + col] = acc[i];
    }
}

// ---------------- K2: attention scores per (node, head) ---------------------
__global__ void gat_attn_scores(const float* __restrict__ xp,
                                const float* __restrict__ aks,   // (C,H)
                                const float* __restrict__ akn,   // (C,H)
                                float* __restrict__ attn_self,
                                float* __restrict__ attn_neigh, int n) {
    const int idx = blockIdx.x * blockDim.x + threadIdx.x;
    if (idx >= n * HEADS) return;
    const int node = idx >> 2, h = idx & 3;
    const float* v = xp + (size_t)node * HC + h * CH;
    float s0 = 0.f, s1 = 0.f;
#pragma unroll
    for (int c = 0; c < CH; ++c) {
        const float t = v[c];
        s0 += t * aks[c * HEADS + h];
        s1 += t * akn[c * HEADS + h];
    }
    attn_self[idx]  = s0;
    attn_neigh[idx] = s1;
}

// ---------------- K3: init out = bias, seg_max = -inf, seg_sum = 0 ----------
__global__ void gat_init(float* __restrict__ out, const float* __restrict__ bias,
                         float* __restrict__ seg_max, float* __restrict__ seg_sum,
                         int n) {
    const int idx = blockIdx.x * blockDim.x + threadIdx.x;
    if (idx < n * HC)    out[idx] = bias[idx & (HC - 1)];
    if (idx < n * HEADS) { seg_max[idx] = -INFINITY; seg_sum[idx] = 0.f; }
}

// ---------------- K4: logits + leakyReLU + segment max ----------------------
__global__ void gat_edge_logits(const int* __restrict__ ei,
                                const float* __restrict__ attn_self,
                                const float* __restrict__ attn_neigh,
                                float* __restrict__ evals,
                                float* __restrict__ seg_max,
                                int ne, int n) {
    const int idx = blockIdx.x * blockDim.x + threadIdx.x;
    const int total = (ne + n) * HEADS;
    if (idx >= total) return;
    const int e = idx >> 2, h = idx & 3;
    int src, tgt;
    if (e < ne) { src = ei[e * 2]; tgt = ei[e * 2 + 1]; }
    else        { src = tgt = e - ne; }                    // self loops
    float l = attn_self[tgt * HEADS + h] + attn_neigh[src * HEADS + h];
    l = (l >= 0.f) ? l : LEAKY * l;
    evals[idx] = l;
    atomicMax(&seg_max[tgt * HEADS + h], l);               // global_atomic_max_num_f32
}

// ---------------- K5: exp + segment sum -------------------------------------
__global__ void gat_edge_exp(const int* __restrict__ ei,
                             float* __restrict__ evals,
                             const float* __restrict__ seg_max,
                             float* __restrict__ seg_sum,
                             int ne, int n) {
    const int idx = blockIdx.x * blockDim.x + threadIdx.x;
    const int total = (ne + n) * HEADS;
    if (idx >= total) return;
    const int e = idx >> 2, h = idx & 3;
    int tgt;
    if (e < ne) tgt = ei[e * 2 + 1]; else tgt = e - ne;
    const float v = __expf(evals[idx] - seg_max[tgt * HEADS + h]);
    evals[idx] = v;
    atomicAdd(&seg_sum[tgt * HEADS + h], v);               // global_atomic_add_f32
}

// ---------------- K6: alpha + message scatter-add ---------------------------
// one thread per (edge, channel); (E+N)*128 threads
__global__ void gat_scatter(const int* __restrict__ ei,
                            const float* __restrict__ evals,
                            const float* __restrict__ seg_sum,
                            const float* __restrict__ xp,
                            float* __restrict__ out,
                            float* __restrict__ alpha_out,
                            int ne, int n) {
    const long long idx = (long long)blockIdx.x * blockDim.x + threadIdx.x;
    const long long total = (long long)(ne + n) * HC;
    if (idx >= total) return;
    const int e  = (int)(idx >> 7);
    const int ch = (int)(idx & (HC - 1));
    const int h  = ch >> 5;
    int src, tgt;
    if (e < ne) { src = ei[e * 2]; tgt = ei[e * 2 + 1]; }
    else        { src = tgt = e - ne; }
    const float alpha = evals[e * HEADS + h] / seg_sum[tgt * HEADS + h];
    if ((ch & (CH - 1)) == 0) alpha_out[e * HEADS + h] = alpha;
    const float msg = alpha * xp[(size_t)src * HC + ch];
    atomicAdd(&out[(size_t)tgt * HC + ch], msg);
}

// ---------------------------------------------------------------------------
extern "C" void kernel_launch(void* const* d_in, const int* in_sizes, int n_in,
                              void* d_out, int out_size, void* d_ws, size_t ws_size,
                              hipStream_t stream) {
    const float* x    = (const float*)d_in[0];
    const int*   ei   = (const int*)d_in[1];          // (E,2) harness int32
    const float* W    = (const float*)d_in[2];        // (F,H,C) == (128,128) row-major
    const float* aks  = (const float*)d_in[3];        // (C,H)
    const float* akn  = (const float*)d_in[4];        // (C,H)
    const float* bias = (const float*)d_in[5];        // (HC,)

    const int n  = in_sizes[0] / HC;                  // 100000
    const int ne = in_sizes[1] / 2;                   // 1600000
    const int tot = ne + n;                           // edges incl. self loops

    // workspace carve-up (floats)
    float* ws       = (float*)d_ws;
    float* xp       = ws;                                  // n*HC
    float* attn_s   = xp + (size_t)n * HC;                 // n*H
    float* attn_n   = attn_s + (size_t)n * HEADS;          // n*H
    float* seg_max  = attn_n + (size_t)n * HEADS;          // n*H
    float* seg_sum  = seg_max + (size_t)n * HEADS;         // n*H
    float* evals    = seg_sum + (size_t)n * HEADS;         // (E+N)*H

    float* out       = (float*)d_out;                      // n*HC
    float* alpha_out = out + (size_t)n * HC;               // (E+N)*H

    // K1: WMMA GEMM. tiles = ceil(n/16) * (HC/16); 4 waves (tiles) per block.
    {
        const int tiles = ((n + 15) / 16) * (HC / 16);
        const int blocks = (tiles + 3) / 4;
        gat_gemm_wmma<<<blocks, 128, 0, stream>>>(x, W, xp, n);
    }
    // K2: attention scores
    {
        const int t = n * HEADS;
        gat_attn_scores<<<(t + 255) / 256, 256, 0, stream>>>(xp, aks, akn, attn_s, attn_n, n);
    }
    // K3: init
    {
        const int t = n * HC;
        gat_init<<<(t + 255) / 256, 256, 0, stream>>>(out, bias, seg_max, seg_sum, n);
    }
    // K4: logits + segment max
    {
        const int t = tot * HEADS;
        gat_edge_logits<<<(t + 255) / 256, 256, 0, stream>>>(ei, attn_s, attn_n, evals, seg_max, ne, n);
    }
    // K5: exp + segment sum
    {
        const int t = tot * HEADS;
        gat_edge_exp<<<(t + 255) / 256, 256, 0, stream>>>(ei, evals, seg_max, seg_sum, ne, n);
    }
    // K6: alpha + scatter
    {
        const long long t = (long long)tot * HC;
        const int blocks = (int)((t + 255) / 256);
        gat_scatter<<<blocks, 256, 0, stream>>>(ei, evals, seg_sum, xp, out, alpha_out, ne, n);
    }
}